// Network_28862180229296
// MI455X (gfx1250) — compile-verified
//
#include <hip/hip_runtime.h>
#include <float.h>

typedef float v2f __attribute__((ext_vector_type(2)));
typedef float v8f __attribute__((ext_vector_type(8)));

// ---------------------------------------------------------------------------
// Dense transform T = X @ W   (X: [N,64], W: [64,64], N % 16 == 0)
// 128 threads = 4 waves; each wave computes one 16x16 output tile of the
// 16-row block using V_WMMA_F32_16X16X4_F32 (16 chained WMMAs over K=64).
// RELU_IN applies relu(x) while staging X into LDS (fuses the previous
// layer's activation, so raw accumulators never need a separate ReLU pass).
// ---------------------------------------------------------------------------
template <bool RELU_IN>
__global__ __launch_bounds__(128) void hmc_gemm(const float* __restrict__ X,
                                                const float* __restrict__ W,
                                                float* __restrict__ T) {
    __shared__ float sX[16 * 64];   // 4 KB  : 16-row tile of X
    __shared__ float sW[64 * 64];   // 16 KB : full weight matrix

    const int tid  = threadIdx.x;
    const int row0 = blockIdx.x * 16;

    // cooperative LDS staging (float4 wide), optional fused ReLU on X
    {
        const float4* xs = (const float4*)(X + (long)row0 * 64);
        float4* xd = (float4*)sX;
#pragma unroll
        for (int i = 0; i < 2; ++i) {
            float4 v = xs[tid + i * 128];
            if (RELU_IN) {
                v.x = fmaxf(v.x, 0.f); v.y = fmaxf(v.y, 0.f);
                v.z = fmaxf(v.z, 0.f); v.w = fmaxf(v.w, 0.f);
            }
            xd[tid + i * 128] = v;
        }
        const float4* wsrc = (const float4*)W;
        float4* wdst = (float4*)sW;
#pragma unroll
        for (int i = 0; i < 8; ++i) wdst[tid + i * 128] = wsrc[tid + i * 128];
    }
    __syncthreads();

    const int wave = tid >> 5;        // 0..3 -> N-tile
    const int lane = tid & 31;
    const int half = lane >> 4;       // 0: K pair {0,1}, 1: K pair {2,3}
    const int l16  = lane & 15;
    const int nt   = wave * 16;

    v8f acc = {};
#pragma unroll
    for (int k0 = 0; k0 < 64; k0 += 4) {
        const int ka = k0 + half * 2;
        v2f a = *(const v2f*)(sX + l16 * 64 + ka);        // A: 16x4 fragment
        v2f b;                                            // B: 4x16 fragment
        b.x = sW[ka * 64 + nt + l16];
        b.y = sW[(ka + 1) * 64 + nt + l16];
        acc = __builtin_amdgcn_wmma_f32_16x16x4_f32(
            false, a, false, b, (short)0, acc, false, false);
    }

    // D layout: VGPR v -> M = v + 8*half; N = nt + l16
#pragma unroll
    for (int v = 0; v < 8; ++v) {
        T[(long)(row0 + half * 8 + v) * 64 + nt + l16] = acc[v];
    }
}

// ---------------------------------------------------------------------------
// Segment-sum without atomics: `rows` is sorted per pair, so each output row
// owns one contiguous edge run. 16 lanes per output row (4 channels each)
// binary-search the run, accumulate gathered T rows in registers, and do a
// single (optionally accumulating) float4 store. Next-edge T row prefetched
// via global_prefetch_b8 to hide the random-gather latency.
// ---------------------------------------------------------------------------
__global__ void row_gather(const float* __restrict__ T,
                           const int* __restrict__ rows,
                           const int* __restrict__ cols,
                           const float* __restrict__ vals,
                           float* __restrict__ H, int E, int Nj,
                           int accumulate) {
    const int tid = blockIdx.x * blockDim.x + threadIdx.x;
    const int j   = tid >> 4;
    const int c4  = (tid & 15) << 2;
    if (j >= Nj) return;

    // lower_bound(rows, j) and upper_bound(rows, j)
    int lo = 0, hi = E;
    while (lo < hi) { int m = (lo + hi) >> 1; if (rows[m] < j) lo = m + 1; else hi = m; }
    const int s = lo;
    hi = E;
    while (lo < hi) { int m = (lo + hi) >> 1; if (rows[m] <= j) lo = m + 1; else hi = m; }
    const int e = lo;

    float4 acc = make_float4(0.f, 0.f, 0.f, 0.f);
    if (s < e) {
        int colq = cols[s];
        for (int q = s; q < e; ++q) {
            const int coln = (q + 1 < e) ? cols[q + 1] : colq;
            __builtin_prefetch(T + (long)coln * 64 + c4, 0, 3);
            const float  v = vals[q];
            const float4 t = *(const float4*)(T + (long)colq * 64 + c4);
            acc.x = fmaf(t.x, v, acc.x);
            acc.y = fmaf(t.y, v, acc.y);
            acc.z = fmaf(t.z, v, acc.z);
            acc.w = fmaf(t.w, v, acc.w);
            colq = coln;
        }
    }

    float4* hp = (float4*)(H + (long)j * 64 + c4);
    if (accumulate) {
        float4 h = *hp;
        acc.x += h.x; acc.y += h.y; acc.z += h.z; acc.w += h.w;
    }
    *hp = acc;
}

// ---------------------------------------------------------------------------
// Pooling stage 1 (coalesced): lanes map to channels, 4 row-groups per block.
// Applies the fused layer-2 epilogue x = relu(h2 + relu(h1)) on the fly.
// Emits per-block partials [sum | sumsq | max | min] (4 x 64 floats).
// ---------------------------------------------------------------------------
__global__ __launch_bounds__(256) void pool_stage1(const float* __restrict__ XA,
                                                   const float* __restrict__ XB,
                                                   int Nrows,
                                                   float* __restrict__ part) {
    const int ch = threadIdx.x & 63;
    const int rg = threadIdx.x >> 6;   // 0..3
    float s = 0.f, s2 = 0.f, mx = -FLT_MAX, mn = FLT_MAX;
    for (int r = blockIdx.x * 4 + rg; r < Nrows; r += gridDim.x * 4) {
        const long idx = (long)r * 64 + ch;
        const float x = fmaxf(XA[idx] + fmaxf(XB[idx], 0.f), 0.f);
        s += x; s2 += x * x;
        mx = fmaxf(mx, x); mn = fminf(mn, x);
    }
    __shared__ float sh[4][4][64];     // [quantity][rowgroup][channel]
    sh[0][rg][ch] = s;  sh[1][rg][ch] = s2;
    sh[2][rg][ch] = mx; sh[3][rg][ch] = mn;
    __syncthreads();
    if (rg == 0) {
        float ts  = sh[0][0][ch] + sh[0][1][ch] + sh[0][2][ch] + sh[0][3][ch];
        float ts2 = sh[1][0][ch] + sh[1][1][ch] + sh[1][2][ch] + sh[1][3][ch];
        float tmx = fmaxf(fmaxf(sh[2][0][ch], sh[2][1][ch]),
                          fmaxf(sh[2][2][ch], sh[2][3][ch]));
        float tmn = fminf(fminf(sh[3][0][ch], sh[3][1][ch]),
                          fminf(sh[3][2][ch], sh[3][3][ch]));
        float* p = part + (long)blockIdx.x * 256;
        p[ch] = ts; p[64 + ch] = ts2; p[128 + ch] = tmx; p[192 + ch] = tmn;
    }
}

// ---------------------------------------------------------------------------
// Pooling stage 2: combine 64 partial sets; emit [mean | std | max | min].
// std is biased (population) with var==0 -> 1e-6, per the reference.
// ---------------------------------------------------------------------------
__global__ __launch_bounds__(64) void pool_stage2(const float* __restrict__ part,
                                                  int Nrows,
                                                  float* __restrict__ out) {
    const int ch = threadIdx.x;
    float s = 0.f, s2 = 0.f, mx = -FLT_MAX, mn = FLT_MAX;
    for (int b = 0; b < 64; ++b) {
        const float* p = part + (long)b * 256;
        s += p[ch]; s2 += p[64 + ch];
        mx = fmaxf(mx, p[128 + ch]); mn = fminf(mn, p[192 + ch]);
    }
    const float inv  = 1.0f / (float)Nrows;
    const float mean = s * inv;
    const float var  = fmaxf(s2 * inv - mean * mean, 0.f);
    const float sd   = sqrtf(var == 0.f ? 1e-6f : var);
    out[ch] = mean; out[64 + ch] = sd; out[128 + ch] = mx; out[192 + ch] = mn;
}

// ---------------------------------------------------------------------------
// Head MLP: 1284 -> 512 -> 128 -> 64 -> 2, ReLU between; out[1] squared.
// ---------------------------------------------------------------------------
__global__ __launch_bounds__(512) void mlp_kernel(
    const float* __restrict__ pooled, const float* __restrict__ gfeat,
    const float* __restrict__ w1, const float* __restrict__ b1,
    const float* __restrict__ w2, const float* __restrict__ b2,
    const float* __restrict__ w3, const float* __restrict__ b3,
    const float* __restrict__ w4, const float* __restrict__ b4,
    float* __restrict__ out) {
    __shared__ float in1[1284];
    __shared__ float h1[512];
    __shared__ float h2[128];
    __shared__ float h3[64];
    const int t = threadIdx.x;
    for (int i = t; i < 1280; i += 512) in1[i] = pooled[i];
    if (t < 4) in1[1280 + t] = gfeat[t];
    __syncthreads();
    {
        float acc = b1[t];
        for (int i = 0; i < 1284; ++i) acc = fmaf(in1[i], w1[i * 512 + t], acc);
        h1[t] = fmaxf(acc, 0.f);
    }
    __syncthreads();
    if (t < 128) {
        float acc = b2[t];
        for (int i = 0; i < 512; ++i) acc = fmaf(h1[i], w2[i * 128 + t], acc);
        h2[t] = fmaxf(acc, 0.f);
    }
    __syncthreads();
    if (t < 64) {
        float acc = b3[t];
        for (int i = 0; i < 128; ++i) acc = fmaf(h2[i], w3[i * 64 + t], acc);
        h3[t] = fmaxf(acc, 0.f);
    }
    __syncthreads();
    if (t < 2) {
        float acc = b4[t];
        for (int i = 0; i < 64; ++i) acc = fmaf(h3[i], w4[i * 2 + t], acc);
        out[t] = (t == 1) ? acc * acc : acc;   // square second half
    }
}

// ---------------------------------------------------------------------------
// Host-side orchestration
// ---------------------------------------------------------------------------
extern "C" void kernel_launch(void* const* d_in, const int* in_sizes, int n_in,
                              void* d_out, int out_size, void* d_ws, size_t ws_size,
                              hipStream_t stream) {
    static const int NRr[5]     = {50000, 100000, 100000, 40000, 20000};
    static const int PAIR_I[15] = {0, 1, 2, 3, 4, 0, 0, 0, 0, 1, 1, 1, 2, 2, 3};
    static const int PAIR_J[15] = {0, 1, 2, 3, 4, 1, 2, 3, 4, 2, 3, 4, 3, 4, 4};

    const float* xin[5];
    for (int r = 0; r < 5; ++r) xin[r] = (const float*)d_in[r];
    const int*   rows  = (const int*)d_in[5];
    const int*   cols  = (const int*)d_in[6];
    const float* vals  = (const float*)d_in[7];
    const float* gfeat = (const float*)d_in[8];
    const float* Whmc  = (const float*)d_in[9];
    const float* w1 = (const float*)d_in[10]; const float* b1 = (const float*)d_in[11];
    const float* w2 = (const float*)d_in[12]; const float* b2 = (const float*)d_in[13];
    const float* w3 = (const float*)d_in[14]; const float* b3 = (const float*)d_in[15];
    const float* w4 = (const float*)d_in[16]; const float* b4 = (const float*)d_in[17];

    float* ws   = (float*)d_ws;
    float* XA   = ws;                  // 19,840,000 floats (raw h, layers 0 and 2)
    float* XB   = ws + 19840000;       // 19,840,000 floats (raw h, layer 1)
    float* Tbuf = ws + 39680000;       //  6,400,000 floats (max N_i * 64)
    float* POOL = ws + 46080000;       //  1,280 floats (+pad)
    float* PART = ws + 46082048;       //  5 * 64 * 256 = 81,920 floats

    long ROFF[5]; long accum = 0;
    for (int r = 0; r < 5; ++r) { ROFF[r] = accum; accum += (long)NRr[r] * 64; }
    long OFF[16]; OFF[0] = 0;
    for (int k = 0; k < 15; ++k) OFF[k + 1] = OFF[k] + 8L * NRr[PAIR_J[k]];

    for (int l = 0; l < 3; ++l) {
        float*       dst     = (l == 1) ? XB : XA;   // l=0 -> XA, l=1 -> XB, l=2 -> XA
        const float* srcBase = (l == 1) ? XA : XB;   // valid for l >= 1

        for (int k = 0; k < 15; ++k) {
            const int i = PAIR_I[k], j = PAIR_J[k];
            const float* Xi  = (l == 0) ? xin[i] : (srcBase + ROFF[i]);
            const float* Wlk = Whmc + ((long)l * 15 + k) * 64 * 64;

            if (l == 0)   // layer-0 inputs are raw features (no activation)
                hmc_gemm<false><<<NRr[i] / 16, 128, 0, stream>>>(Xi, Wlk, Tbuf);
            else          // fuse relu(prev layer h) into the LDS staging
                hmc_gemm<true><<<NRr[i] / 16, 128, 0, stream>>>(Xi, Wlk, Tbuf);

            const int E = (int)(OFF[k + 1] - OFF[k]);
            // first pair targeting each rank (k < 5) initializes; rest accumulate
            row_gather<<<NRr[j] / 16, 256, 0, stream>>>(
                Tbuf, rows + OFF[k], cols + OFF[k], vals + OFF[k],
                dst + ROFF[j], E, NRr[j], (k < 5) ? 0 : 1);
        }
        // No standalone ReLU pass: layers 1/2 consume relu() inside hmc_gemm
        // staging; the layer-2 epilogue is fused into pool_stage1.
    }

    for (int r = 0; r < 5; ++r) {
        pool_stage1<<<64, 256, 0, stream>>>(XA + ROFF[r], XB + ROFF[r], NRr[r],
                                            PART + (long)r * 64 * 256);
        pool_stage2<<<1, 64, 0, stream>>>(PART + (long)r * 64 * 256, NRr[r],
                                          POOL + (long)r * 256);
    }

    mlp_kernel<<<1, 512, 0, stream>>>(POOL, gfeat, w1, b1, w2, b2, w3, b3, w4, b4,
                                      (float*)d_out);
}